// SelfAttention_44925357916178
// MI455X (gfx1250) — compile-verified
//
#include <hip/hip_runtime.h>
#include <math.h>

#define HID    1024
#define NHEADS 16
#define HDIM   64
#define KVLEN  4096

typedef __attribute__((ext_vector_type(2))) float v2f;
typedef __attribute__((ext_vector_type(4))) float v4f;
typedef __attribute__((ext_vector_type(8))) float v8f;

// ---------------------------------------------------------------------------
// Kernel 1: flash-style single-query attention, split-softmax (barrier-free
// main loop). grid = 512 (one block per (b,h)), block = 256 threads (8 waves).
// Group g = tid>>2 (64 groups) owns key rows {g, g+64, ..., g+4032} and keeps
// a private online-softmax state (m_g, l_g, acc). Lane j = tid&3 owns
// head-dims [j*16, j*16+16). One merge through LDS at the end.
// ---------------------------------------------------------------------------
__global__ __launch_bounds__(256) void attn_flash_kernel(
    const float* __restrict__ q,
    const float* __restrict__ key,
    const float* __restrict__ val,
    float* __restrict__ ctx)
{
    __shared__ float s_q[HDIM];
    __shared__ float s_m[64];
    __shared__ float s_l[64];
    __shared__ float s_acc[64 * HDIM];   // 16 KB

    const int bh  = blockIdx.x;
    const int b   = bh >> 4;
    const int h   = bh & 15;
    const int tid = threadIdx.x;
    const int g   = tid >> 2;
    const int j   = tid & 3;

    if (tid < HDIM) s_q[tid] = q[(size_t)b * HID + h * HDIM + tid];
    __syncthreads();

    float qv[16];
#pragma unroll
    for (int i = 0; i < 16; ++i) qv[i] = s_q[j * 16 + i];

    const float inv_scale = 0.125f;      // 1/sqrt(64)
    float m_run = -INFINITY;
    float l_run = 0.f;
    float acc[16];
#pragma unroll
    for (int i = 0; i < 16; ++i) acc[i] = 0.f;

    const size_t kv_base = (size_t)b * KVLEN * HID + (size_t)h * HDIM;

    // -------- barrier-free streaming loop: 64 rows/iter across the block ----
    for (int c = 0; c < KVLEN / 64; ++c) {
        const int row = c * 64 + g;
        const v4f* kp = (const v4f*)(key + kv_base + (size_t)row * HID + j * 16);
        const v4f* vp = (const v4f*)(val + kv_base + (size_t)row * HID + j * 16);

        // issue all K and V loads up front (non-temporal: single-use stream)
        v4f kk[4], vv[4];
#pragma unroll
        for (int i = 0; i < 4; ++i) kk[i] = __builtin_nontemporal_load(&kp[i]);
#pragma unroll
        for (int i = 0; i < 4; ++i) vv[i] = __builtin_nontemporal_load(&vp[i]);

        // q . k (16 dims per lane, reduce across the 4 lanes of the group)
        float dot = 0.f;
#pragma unroll
        for (int i = 0; i < 4; ++i) {
            dot += qv[i * 4 + 0] * kk[i].x + qv[i * 4 + 1] * kk[i].y
                 + qv[i * 4 + 2] * kk[i].z + qv[i * 4 + 3] * kk[i].w;
        }
        dot += __shfl_xor(dot, 1, 4);
        dot += __shfl_xor(dot, 2, 4);
        const float score = dot * inv_scale;   // group-uniform

        // per-group online softmax (no cross-group communication)
        const float m_new   = fmaxf(m_run, score);
        const float rescale = __expf(m_run - m_new);
        const float p       = __expf(score - m_new);
        l_run = l_run * rescale + p;
        m_run = m_new;

#pragma unroll
        for (int i = 0; i < 4; ++i) {
            acc[i * 4 + 0] = acc[i * 4 + 0] * rescale + p * vv[i].x;
            acc[i * 4 + 1] = acc[i * 4 + 1] * rescale + p * vv[i].y;
            acc[i * 4 + 2] = acc[i * 4 + 2] * rescale + p * vv[i].z;
            acc[i * 4 + 3] = acc[i * 4 + 3] * rescale + p * vv[i].w;
        }
    }

    // -------- merge the 64 per-group partial softmax states -----------------
    if (j == 0) { s_m[g] = m_run; s_l[g] = l_run; }
    __syncthreads();

    float M = -INFINITY;
#pragma unroll 8
    for (int gg = 0; gg < 64; ++gg) M = fmaxf(M, s_m[gg]);
    float L = 0.f;
#pragma unroll 8
    for (int gg = 0; gg < 64; ++gg) L += s_l[gg] * __expf(s_m[gg] - M);

    const float sc = __expf(m_run - M);
#pragma unroll
    for (int i = 0; i < 16; ++i) s_acc[g * HDIM + j * 16 + i] = acc[i] * sc;
    __syncthreads();

    if (tid < HDIM) {
        float s = 0.f;
        for (int gg = 0; gg < 64; ++gg) s += s_acc[gg * HDIM + tid];
        ctx[(size_t)b * HID + h * HDIM + tid] = s / L;
    }
}

// ---------------------------------------------------------------------------
// Kernel 2: out = ctx @ fc_w^T + fc_b  via V_WMMA_F32_16X16X4_F32.
// grid = (8,2), block = 256 (8 waves). Each wave owns one 16x16 output tile:
// mtile = blockIdx.y (M=32 -> 2 tiles), ntile = blockIdx.x*8 + wave (64 tiles).
// A layout: lane l<16 -> M=l, K pair {k,k+1}; lanes 16-31 -> K pair {k+2,k+3}.
// B layout mirrors with N = lane&15. D: VGPR i, lanes<16 -> M=i, lanes>=16 -> M=i+8.
// ---------------------------------------------------------------------------
__global__ __launch_bounds__(256) void fc_wmma_kernel(
    const float* __restrict__ ctx,
    const float* __restrict__ w,      // (HID, HID), out[m,n] = sum_k ctx[m,k]*w[n,k]
    const float* __restrict__ bias,
    float* __restrict__ out)
{
    const int lane  = threadIdx.x & 31;
    const int wave  = threadIdx.x >> 5;
    const int ntile = blockIdx.x * 8 + wave;   // 0..63
    const int mtile = blockIdx.y;              // 0..1
    const int m0 = mtile * 16;
    const int n0 = ntile * 16;
    const int r  = lane & 15;
    const int kh = (lane >> 4) * 2;            // 0 or 2

    const float* arow = ctx + (size_t)(m0 + r) * HID;   // A row M=r
    const float* brow = w   + (size_t)(n0 + r) * HID;   // B col N=r (w row-major over n)

    v8f c = {};
    for (int k = 0; k < HID; k += 4) {
        v2f a  = *(const v2f*)(arow + k + kh);
        v2f bb = *(const v2f*)(brow + k + kh);
        // 8 args: (neg_a, A, neg_b, B, c_mod, C, reuse_a, reuse_b)
        c = __builtin_amdgcn_wmma_f32_16x16x4_f32(
                false, a, false, bb, (short)0, c, false, false);
    }

    const int col   = n0 + r;
    const int mbase = m0 + (lane >> 4) * 8;
    const float bval = bias[col];
#pragma unroll
    for (int i = 0; i < 8; ++i)
        out[(size_t)(mbase + i) * HID + col] = c[i] + bval;
}

// ---------------------------------------------------------------------------
extern "C" void kernel_launch(void* const* d_in, const int* in_sizes, int n_in,
                              void* d_out, int out_size, void* d_ws, size_t ws_size,
                              hipStream_t stream) {
    const float* q   = (const float*)d_in[0];   // (32, 1024)
    const float* key = (const float*)d_in[1];   // (32, 4096, 1024)
    const float* val = (const float*)d_in[2];   // (32, 4096, 1024)
    const float* fcw = (const float*)d_in[3];   // (1024, 1024)
    const float* fcb = (const float*)d_in[4];   // (1024,)
    float* out = (float*)d_out;                 // (32, 1024)
    float* ctx = (float*)d_ws;                  // 32*1024 floats scratch

    attn_flash_kernel<<<dim3(32 * NHEADS), dim3(256), 0, stream>>>(q, key, val, ctx);
    fc_wmma_kernel<<<dim3(8, 2), dim3(256), 0, stream>>>(ctx, fcw, fcb, out);
}